// TrafficAugmentation_50714973831499
// MI455X (gfx1250) — compile-verified
//
#include <hip/hip_runtime.h>

#define SS 4096
#define MSSF 1448.0f

typedef __attribute__((ext_vector_type(4))) unsigned v4u;
typedef __attribute__((ext_vector_type(8))) unsigned v8u;
typedef __attribute__((address_space(3))) char lds_char_t;

// Byte offset of a __shared__ object within the workgroup LDS allocation
// (addrspacecast generic -> LDS, then ptrtoint of the 32-bit LDS pointer).
static __device__ __forceinline__ unsigned lds_off(void* p) {
    return (unsigned)(unsigned long long)(lds_char_t*)p;
}

// One Tensor Data Mover DMA: contiguous row of `nelem` fp32 from global -> LDS.
// D# group0/group1 per CDNA5 ISA 8.3/8.4 (count=1, type=2 "image", data_size=4B,
// 1-D tile: tensor_dim0 = tile_dim0 = nelem, tensor_dim1 = tile_dim1 = 1).
static __device__ __forceinline__ void tdm_load_row(const float* g, unsigned lds_byte,
                                                    unsigned nelem) {
    unsigned long long ga = (unsigned long long)g;
    v4u g0;
    g0[0] = 1u;                                             // count=1, user descriptor
    g0[1] = lds_byte;                                       // lds_addr
    g0[2] = (unsigned)(ga & 0xFFFFFFFFull);                 // global_addr[31:0]
    g0[3] = (unsigned)((ga >> 32) & 0x1FFFFFFull)           // global_addr[56:32]
          | (2u << 30);                                     // type=2
    v8u g1;
    g1[0] = 2u << 16;                                       // data_size=2 (4 bytes)
    g1[1] = (nelem & 0xFFFFu) << 16;                        // tensor_dim0[15:0]
    g1[2] = ((nelem >> 16) & 0xFFFFu) | (1u << 16);         // tensor_dim0[31:16], tensor_dim1=1
    g1[3] = (nelem & 0xFFFFu) << 16;                        // tile_dim0
    g1[4] = 1u;                                             // tile_dim1=1, tile_dim2=0
    g1[5] = nelem;                                          // tensor_dim0_stride[31:0]
    g1[6] = (nelem & 0xFFFFu) << 16;                        // stride0[47:32]=0, stride1[15:0]
    g1[7] = 0u;                                             // stride1[47:16]
    asm volatile("tensor_load_to_lds %0, %1" :: "s"(g0), "s"(g1) : "memory");
}

__global__ __launch_bounds__(32)
void TrafficAugmentation_50714973831499_kernel(const float* __restrict__ x,
                                               const float* __restrict__ dl,
                                               const float* __restrict__ rt,
                                               float* __restrict__ out) {
    __shared__ float s_x[SS];            // packet lengths
    __shared__ float s_d[SS];            // delays   (reused: rec_start as int)
    __shared__ float s_r[SS];            // rtts     (reused: rec_rem)
    __shared__ float s_buf[SS];          // burst byte sums (reused: rec_nfull as int)
    __shared__ unsigned short s_nxt[SS]; // burst-start -> next burst-start
    __shared__ int s_nb;
    __shared__ int s_total;

    const int row  = blockIdx.x;
    const int lane = threadIdx.x;
    const long long base = (long long)row * SS;

    // --- Stage the row into LDS via TDM (3 x 16KB DMA, overlapped, one wait) ---
    tdm_load_row(x  + base, lds_off(s_x), SS);
    tdm_load_row(dl + base, lds_off(s_d), SS);
    tdm_load_row(rt + base, lds_off(s_r), SS);
    __builtin_amdgcn_s_wait_tensorcnt(0);
    __syncthreads();

    // --- Phase A: for every candidate start j, simulate its burst exactly ---
    // (same fp32 op order as the reference: r -= d[k]; buf += x[k]; stop at r<=0)
    for (int j = lane; j < SS; j += 32) {
        float r   = s_r[j];
        float acc = 0.0f;
        int   k   = j;
        while (k < SS) {
            r   -= s_d[k];
            acc += s_x[k];
            ++k;
            if (r <= 0.0f) break;
        }
        s_nxt[j] = (unsigned short)k;   // k = end+1 (or SS)
        s_buf[j] = acc;
    }
    __syncthreads();

    // --- Phase B: lane 0 walks the burst chain, emits burst records ---
    if (lane == 0) {
        // Aliased record arrays are safe: record index nb is strictly less
        // than every future read index `start` (each burst consumes >=1 pkt).
        int*   rec_nfull = (int*)s_buf;
        float* rec_rem   = s_r;
        int*   rec_start = (int*)s_d;
        int start = 0, nb = 0, opos = 0;
        while (start < SS) {
            if (!(s_x[start] > 0.0f)) break;           // outer-while stop
            float buf    = s_buf[start];
            int   nstart = s_nxt[start];
            int   nfull  = (int)__builtin_ceilf(buf / MSSF) - 1;
            if (nfull < 0) nfull = 0;
            float rem = buf - (float)nfull * MSSF;
            int   cnt = nfull + (rem > 0.0f ? 1 : 0);
            rec_nfull[nb] = nfull;
            rec_rem[nb]   = rem;
            rec_start[nb] = opos;
            ++nb;
            opos += cnt;
            start = nstart;
        }
        s_nb    = nb;
        s_total = (opos < SS) ? opos : SS;
    }
    __syncthreads();

    // --- Phase C: parallel emission (MSS runs + remainders) and zero tail ---
    const int nb    = s_nb;
    const int total = s_total;
    float* orow = out + base;

    for (int p = total + lane; p < SS; p += 32)   // coalesced zero tail
        orow[p] = 0.0f;

    const int*   rec_nfull = (const int*)s_buf;
    const float* rec_rem   = s_r;
    const int*   rec_start = (const int*)s_d;
    for (int b = lane; b < nb; b += 32) {
        int   s0  = rec_start[b];
        int   nf  = rec_nfull[b];
        float rem = rec_rem[b];
        int e0 = s0 + nf; if (e0 > SS) e0 = SS;
        for (int p = s0; p < e0; ++p) orow[p] = MSSF;   // full segments
        int rp = s0 + nf;
        if (rem > 0.0f && rp < SS) orow[rp] = rem;       // remainder (truncation-safe)
    }
}

extern "C" void kernel_launch(void* const* d_in, const int* in_sizes, int n_in,
                              void* d_out, int out_size, void* d_ws, size_t ws_size,
                              hipStream_t stream) {
    const float* x  = (const float*)d_in[0];
    const float* dl = (const float*)d_in[1];
    const float* rt = (const float*)d_in[2];
    float* out = (float*)d_out;
    const int rows = in_sizes[0] / SS;   // B = 2048
    hipLaunchKernelGGL(TrafficAugmentation_50714973831499_kernel,
                       dim3(rows), dim3(32), 0, stream, x, dl, rt, out);
}